// BIMPM_49409303773252
// MI455X (gfx1250) — compile-verified
//
#include <hip/hip_runtime.h>
#include <math.h>
#include <stdint.h>

// ---- problem dims (match reference) ----
#define BB    32      // batch
#define SEQ   64      // SA == SB
#define HD    128     // H
#define WDIM  300     // word dim
#define LP    20      // perspectives
#define GD    512     // 4H gates
#define MVD   160     // 8L match vector
#define NCLS  3
#define HCAT  512     // 4H concat of hidden states
#define FC1D  256     // 2H

typedef __attribute__((ext_vector_type(2))) float v2f;
typedef __attribute__((ext_vector_type(8))) float v8f;
typedef __attribute__((ext_vector_type(4))) unsigned int v4u;
typedef __attribute__((ext_vector_type(8))) int v8i;
typedef __attribute__((ext_vector_type(4))) int v4i;

__device__ __forceinline__ float sigm(float x) { return 1.0f / (1.0f + __expf(-x)); }
__device__ __forceinline__ float safeden(float d) { return d > 1e-8f ? d : 1e-8f; }

// ---------------------------------------------------------------------------
// TDM: async DMA of a 2D fp32 tile (tile_dim0=GD contiguous cols, tile_dim1=BB
// rows, row stride SEQ*GD elements) from global into LDS. Tracked by TENSORcnt.
// Descriptor per CDNA5 ISA.
// ---------------------------------------------------------------------------
__device__ __forceinline__ void tdm_load_gate_tile(const float* gptr, unsigned lds_off)
{
    unsigned long long ga = (unsigned long long)(uintptr_t)gptr;
    v4u g0;
    g0.x = 1u;                                       // count=1 (valid), user mode
    g0.y = lds_off;                                  // lds_addr (bytes)
    g0.z = (unsigned)(ga & 0xFFFFFFFFu);             // global_addr[31:0]
    g0.w = (unsigned)((ga >> 32) & 0x01FFFFFFu)      // global_addr[56:32]
         | 0x80000000u;                              // type=2 ("image") at [127:126]
    v8i g1;
    g1[0] = 0x00020000;                              // data_size=2 -> 4 bytes
    g1[1] = 0;                                       // tensor_dim0[15:0]<<16 (1<<30 -> 0)
    g1[2] = 0x00004000;                              // tensor_dim0[31:16]=0x4000; dim1 lo=0
    g1[3] = 0x00004000 | (GD << 16);                 // tensor_dim1[31:16]; tile_dim0=512
    g1[4] = BB;                                      // tile_dim1=32, tile_dim2=0
    g1[5] = SEQ * GD;                                // tensor_dim0_stride[31:0] (elements)
    g1[6] = 0;                                       // stride0 hi / stride1 lo
    g1[7] = 0;                                       // stride1 hi
    v4i z4 = {0, 0, 0, 0};
#if defined(__clang_major__) && (__clang_major__ >= 23)
    v8i z8 = {0, 0, 0, 0, 0, 0, 0, 0};
    __builtin_amdgcn_tensor_load_to_lds(g0, g1, z4, z4, z8, 0);
#else
    __builtin_amdgcn_tensor_load_to_lds(g0, g1, z4, z4, 0);
#endif
}

// ---------------------------------------------------------------------------
// Embedding gather
// ---------------------------------------------------------------------------
__global__ void k_gather(const int* __restrict__ tok, const float* __restrict__ emb,
                         float* __restrict__ out, int nrows)
{
    int idx = blockIdx.x * blockDim.x + threadIdx.x;
    int total = nrows * WDIM;
    if (idx >= total) return;
    int row = idx / WDIM, d = idx % WDIM;
    out[idx] = emb[(long)tok[row] * WDIM + d];
}

// ---------------------------------------------------------------------------
// WMMA GEMM: C[M,N] = X[M,K] @ W[N,K]^T + b1 + b2  (optional tanh)
// fp32 WMMA 16x16x4, one 16x16 output tile per wave. M,N mult of 16, K mult of 4.
// ---------------------------------------------------------------------------
__global__ void k_gemm_bt_wmma(const float* __restrict__ X, const float* __restrict__ W,
                               const float* __restrict__ b1, const float* __restrict__ b2,
                               float* __restrict__ C, int M, int N, int K, int act)
{
    int gwave  = (blockIdx.x * blockDim.x + threadIdx.x) >> 5;
    int lane   = threadIdx.x & 31;
    int tilesN = N >> 4;
    int tiles  = (M >> 4) * tilesN;
    if (gwave >= tiles) return;
    int tm = gwave / tilesN, tn = gwave % tilesN;
    int l15 = lane & 15;
    bool hi = lane >= 16;
    int mrow = tm * 16 + l15;
    int ncol = tn * 16 + l15;
    const float* xr = X + (long)mrow * K;
    const float* wr = W + (long)ncol * K;
    v8f acc = {0.f,0.f,0.f,0.f,0.f,0.f,0.f,0.f};
    for (int k = 0; k < K; k += 4) {
        int ka = k + (hi ? 2 : 0);
        v2f a, b;
        a.x = xr[ka]; a.y = xr[ka + 1];
        b.x = wr[ka]; b.y = wr[ka + 1];
        acc = __builtin_amdgcn_wmma_f32_16x16x4_f32(false, a, false, b,
                                                    (short)0, acc, false, false);
    }
    int col = tn * 16 + l15;
    float bias = (b1 ? b1[col] : 0.f) + (b2 ? b2[col] : 0.f);
    for (int r = 0; r < 8; ++r) {
        int row = tm * 16 + r + (hi ? 8 : 0);
        float v = acc[r] + bias;
        if (act == 1) v = tanhf(v);
        C[(long)row * N + col] = v;
    }
}

// ---------------------------------------------------------------------------
// LSTM recurrence: 4 independent (sequence, direction) jobs, one block each.
// h,c live in LDS. Gate inputs Xpre[t] (32x512 tile, 64KB) are DMA'd into LDS
// by the Tensor Data Mover one timestep ahead (double-buffered), so the L2->LDS
// transfer overlaps the recurrent WMMA GEMM instead of sitting on the serial
// dependency chain. Per step: gates = h @ Whh^T (fp32 WMMA, 32x512x128)
// accumulated in-place into the staged tile, then elementwise state update.
// backward=1: iterate t = S-1..0, store h at position t.
// LDS: 16KB h + 16KB c + 2*64KB staged tiles = 160KB (<320KB/WGP).
// ---------------------------------------------------------------------------
struct LstmJob  { const float* X; const float* Whh; float* Out; int backward; };
struct LstmJobs { LstmJob j[4]; };

__global__ void __launch_bounds__(256) k_lstm4(LstmJobs jobs)
{
    __shared__ float sh[BB * HD];        // 16 KB
    __shared__ float sc[BB * HD];        // 16 KB
    __shared__ float sx[2][BB * GD];     // 2 x 64 KB staged gate-input tiles
    LstmJob jb = jobs.j[blockIdx.x];
    int tid  = threadIdx.x;
    int lane = tid & 31;
    int wave = tid >> 5;                 // 0..7
    int l15  = lane & 15;
    bool hi  = lane >= 16;

    for (int i = tid; i < BB * HD; i += blockDim.x) { sh[i] = 0.f; sc[i] = 0.f; }

    int t0 = jb.backward ? (SEQ - 1) : 0;
    if (wave == 0) {
        // prefetch first timestep's gate inputs
        tdm_load_gate_tile(jb.X + (size_t)t0 * GD, (unsigned)(uintptr_t)&sx[0][0]);
    }
    __syncthreads();

    for (int step = 0; step < SEQ; ++step) {
        int t   = jb.backward ? (SEQ - 1 - step) : step;
        int buf = step & 1;
        float* sxb = &sx[buf][0];

        // issue prefetch of next timestep into the other buffer, then wait for
        // the current buffer's DMA (TDM ops from one wave complete in order).
        if (wave == 0) {
            if (step + 1 < SEQ) {
                int tn1 = jb.backward ? (t - 1) : (t + 1);
                tdm_load_gate_tile(jb.X + (size_t)tn1 * GD,
                                   (unsigned)(uintptr_t)&sx[buf ^ 1][0]);
                __builtin_amdgcn_s_wait_tensorcnt((short)1);
            } else {
                __builtin_amdgcn_s_wait_tensorcnt((short)0);
            }
        }
        __syncthreads();   // staged tile sx[buf] now valid for all waves

        // gates(32x512) += h(32x128) @ Whh(512x128)^T : 2 M-tiles x 32 N-tiles,
        // 8 waves x (2x4) tiles each; accumulate in place into staged tile.
        for (int tm = 0; tm < 2; ++tm) {
            for (int tt = 0; tt < 4; ++tt) {
                int tn   = wave * 4 + tt;
                int mrow = tm * 16 + l15;
                int ncol = tn * 16 + l15;
                const float* hrow = sh + mrow * HD;
                const float* wr   = jb.Whh + (long)ncol * HD;
                v8f acc = {0.f,0.f,0.f,0.f,0.f,0.f,0.f,0.f};
                for (int k = 0; k < HD; k += 4) {
                    int ka = k + (hi ? 2 : 0);
                    v2f a, b;
                    a.x = hrow[ka]; a.y = hrow[ka + 1];
                    b.x = wr[ka];   b.y = wr[ka + 1];
                    acc = __builtin_amdgcn_wmma_f32_16x16x4_f32(false, a, false, b,
                                                                (short)0, acc, false, false);
                }
                int col = tn * 16 + l15;
                for (int r = 0; r < 8; ++r) {
                    int row = tm * 16 + r + (hi ? 8 : 0);
                    int idx = row * GD + col;
                    sxb[idx] = acc[r] + sxb[idx];   // in-place: one thread per idx
                }
            }
        }
        __syncthreads();

        // gate nonlinearity + state update (i,f,g,o order)
        for (int e = tid; e < BB * HD; e += blockDim.x) {
            int bm = e / HD, d = e % HD;
            float gi = sxb[bm * GD + d];
            float gf = sxb[bm * GD + HD + d];
            float gg = sxb[bm * GD + 2 * HD + d];
            float go = sxb[bm * GD + 3 * HD + d];
            float c  = sigm(gf) * sc[e] + sigm(gi) * tanhf(gg);
            float h  = sigm(go) * tanhf(c);
            sc[e] = c; sh[e] = h;
            jb.Out[((long)bm * SEQ + t) * HD + d] = h;
        }
        __syncthreads();
    }
}

// ---------------------------------------------------------------------------
// Row L2 norms: out[b*S+t] = ||v[b,t,:]||
// ---------------------------------------------------------------------------
__global__ void k_norm(const float* __restrict__ v, float* __restrict__ out)
{
    int idx = blockIdx.x * blockDim.x + threadIdx.x;
    if (idx >= BB * SEQ) return;
    const float* p = v + (long)idx * HD;
    float s = 0.f;
    for (int d = 0; d < HD; ++d) s += p[d] * p[d];
    out[idx] = sqrtf(s);
}

// ---------------------------------------------------------------------------
// Attention: att[b,i,j] = cos(v1[b,i], v2[b,j]) with _div semantics
// ---------------------------------------------------------------------------
__global__ void k_attention(const float* __restrict__ v1, const float* __restrict__ v2,
                            const float* __restrict__ n1, const float* __restrict__ n2,
                            float* __restrict__ att)
{
    int idx = blockIdx.x * blockDim.x + threadIdx.x;
    if (idx >= BB * SEQ * SEQ) return;
    int j = idx % SEQ; int i = (idx / SEQ) % SEQ; int b = idx / (SEQ * SEQ);
    const float* p1 = v1 + ((long)b * SEQ + i) * HD;
    const float* p2 = v2 + ((long)b * SEQ + j) * HD;
    float s = 0.f;
    for (int d = 0; d < HD; ++d) s += p1[d] * p2[d];
    att[idx] = s / safeden(n1[b * SEQ + i] * n2[b * SEQ + j]);
}

// ---------------------------------------------------------------------------
// Attentive mean + max (transposed=1 -> reduce over first att index)
// ---------------------------------------------------------------------------
__global__ void k_attn_combine(const float* __restrict__ att, const float* __restrict__ v,
                               float* __restrict__ meanOut, float* __restrict__ maxOut,
                               int transposed)
{
    int idx = blockIdx.x * blockDim.x + threadIdx.x;
    if (idx >= BB * SEQ * HD) return;
    int d = idx % HD; int p = (idx / HD) % SEQ; int b = idx / (HD * SEQ);
    float s = 0.f, ssum = 0.f, mx = -3.4e38f;
    for (int q = 0; q < SEQ; ++q) {
        float a  = transposed ? att[((long)b * SEQ + q) * SEQ + p]
                              : att[((long)b * SEQ + p) * SEQ + q];
        float vv = v[((long)b * SEQ + q) * HD + d];
        s += a * vv; ssum += a;
        mx = fmaxf(mx, vv * a);
    }
    meanOut[idx] = s / safeden(ssum);
    maxOut[idx]  = mx;
}

// ---------------------------------------------------------------------------
// Full match: out[b,t, outOff+l] = cos_w_l(v1[b,t], v2[...])
// v2 addressed as v2 + b*v2_bs + t*v2_ts (v2_ts=0 -> per-batch vector)
// ---------------------------------------------------------------------------
__global__ void k_full_match(const float* __restrict__ v1, const float* __restrict__ v2,
                             int v2_bs, int v2_ts, const float* __restrict__ w,
                             float* __restrict__ out, int outOff)
{
    int idx = blockIdx.x * blockDim.x + threadIdx.x;
    if (idx >= BB * SEQ * LP) return;
    int l = idx % LP; int t = (idx / LP) % SEQ; int b = idx / (LP * SEQ);
    const float* p1 = v1 + ((long)b * SEQ + t) * HD;
    const float* p2 = v2 + (long)b * v2_bs + (long)t * v2_ts;
    const float* pw = w + l * HD;
    float num = 0.f, n1 = 0.f, n2 = 0.f;
    for (int d = 0; d < HD; ++d) {
        float ww = pw[d];
        float a = p1[d] * ww, c = p2[d] * ww;
        num += a * c; n1 += a * a; n2 += c * c;
    }
    out[((long)b * SEQ + t) * MVD + outOff + l] = num / safeden(sqrtf(n1) * sqrtf(n2));
}

// ---------------------------------------------------------------------------
// Maxpool match: out[b,t,outOff+l] = max_j cos_w_l(v1[b,t], v2[b,j])
// ---------------------------------------------------------------------------
__global__ void k_maxpool_match(const float* __restrict__ v1, const float* __restrict__ v2,
                                const float* __restrict__ w, float* __restrict__ out,
                                int outOff)
{
    int idx = blockIdx.x * blockDim.x + threadIdx.x;
    if (idx >= BB * SEQ * LP) return;
    int l = idx % LP; int t = (idx / LP) % SEQ; int b = idx / (LP * SEQ);
    const float* p1 = v1 + ((long)b * SEQ + t) * HD;
    const float* pw = w + l * HD;
    float n1 = 0.f;
    for (int d = 0; d < HD; ++d) { float a = p1[d] * pw[d]; n1 += a * a; }
    n1 = sqrtf(n1);
    float best = -3.4e38f;
    for (int j = 0; j < SEQ; ++j) {
        const float* p2 = v2 + ((long)b * SEQ + j) * HD;
        float num = 0.f, n2 = 0.f;
        for (int d = 0; d < HD; ++d) {
            float ww = pw[d];
            float c = p2[d] * ww;
            num += p1[d] * ww * c; n2 += c * c;
        }
        best = fmaxf(best, num / safeden(n1 * sqrtf(n2)));
    }
    out[((long)b * SEQ + t) * MVD + outOff + l] = best;
}

// ---------------------------------------------------------------------------
// Concat final hiddens: [hAf (t=S-1), hAb (t=0), hBf (t=S-1), hBb (t=0)]
// ---------------------------------------------------------------------------
__global__ void k_hcat(const float* __restrict__ aAf, const float* __restrict__ aAb,
                       const float* __restrict__ aBf, const float* __restrict__ aBb,
                       float* __restrict__ hcat)
{
    int idx = blockIdx.x * blockDim.x + threadIdx.x;
    if (idx >= BB * HCAT) return;
    int n = idx % HCAT; int b = idx / HCAT;
    float v;
    if (n < HD)          v = aAf[((long)b * SEQ + SEQ - 1) * HD + n];
    else if (n < 2 * HD) v = aAb[((long)b * SEQ) * HD + (n - HD)];
    else if (n < 3 * HD) v = aBf[((long)b * SEQ + SEQ - 1) * HD + (n - 2 * HD)];
    else                 v = aBb[((long)b * SEQ) * HD + (n - 3 * HD)];
    hcat[idx] = v;
}

// ---------------------------------------------------------------------------
// fc2 + softmax (N=3 too small for WMMA): one thread per batch row
// ---------------------------------------------------------------------------
__global__ void k_fc2_softmax(const float* __restrict__ x, const float* __restrict__ W,
                              const float* __restrict__ bias, float* __restrict__ out)
{
    int b = blockIdx.x * blockDim.x + threadIdx.x;
    if (b >= BB) return;
    float lg[NCLS];
    float mx = -3.4e38f;
    for (int c = 0; c < NCLS; ++c) {
        float s = bias[c];
        for (int k = 0; k < FC1D; ++k) s += x[b * FC1D + k] * W[c * FC1D + k];
        lg[c] = s; mx = fmaxf(mx, s);
    }
    float sum = 0.f;
    for (int c = 0; c < NCLS; ++c) { lg[c] = __expf(lg[c] - mx); sum += lg[c]; }
    for (int c = 0; c < NCLS; ++c) out[b * NCLS + c] = lg[c] / sum;
}

// ---------------------------------------------------------------------------
// Host launcher
// ---------------------------------------------------------------------------
extern "C" void kernel_launch(void* const* d_in, const int* in_sizes, int n_in,
                              void* d_out, int out_size, void* d_ws, size_t ws_size,
                              hipStream_t stream)
{
    (void)in_sizes; (void)n_in; (void)out_size; (void)ws_size;

    const int*   Atok      = (const int*)d_in[0];
    const int*   Btok      = (const int*)d_in[1];
    const float* emb       = (const float*)d_in[2];
    const float* ctx_Wih_f = (const float*)d_in[3];
    const float* ctx_Whh_f = (const float*)d_in[4];
    const float* ctx_bih_f = (const float*)d_in[5];
    const float* ctx_bhh_f = (const float*)d_in[6];
    const float* agg_Wih_f = (const float*)d_in[7];
    const float* agg_Whh_f = (const float*)d_in[8];
    const float* agg_bih_f = (const float*)d_in[9];
    const float* agg_bhh_f = (const float*)d_in[10];
    const float* ctx_Wih_b = (const float*)d_in[11];
    const float* ctx_Whh_b = (const float*)d_in[12];
    const float* ctx_bih_b = (const float*)d_in[13];
    const float* ctx_bhh_b = (const float*)d_in[14];
    const float* agg_Wih_b = (const float*)d_in[15];
    const float* agg_Whh_b = (const float*)d_in[16];
    const float* agg_bih_b = (const float*)d_in[17];
    const float* agg_bhh_b = (const float*)d_in[18];
    const float* mp_w1 = (const float*)d_in[19];
    const float* mp_w2 = (const float*)d_in[20];
    const float* mp_w3 = (const float*)d_in[21];
    const float* mp_w4 = (const float*)d_in[22];
    const float* mp_w5 = (const float*)d_in[23];
    const float* mp_w6 = (const float*)d_in[24];
    const float* mp_w7 = (const float*)d_in[25];
    const float* mp_w8 = (const float*)d_in[26];
    const float* fc1_W = (const float*)d_in[27];
    const float* fc1_b = (const float*)d_in[28];
    const float* fc2_W = (const float*)d_in[29];
    const float* fc2_b = (const float*)d_in[30];

    float* ws = (float*)d_ws;
    const size_t SZ_EMB = (size_t)BB * SEQ * WDIM;
    const size_t SZ_X   = (size_t)BB * SEQ * GD;
    const size_t SZ_C   = (size_t)BB * SEQ * HD;
    const size_t SZ_N   = (size_t)BB * SEQ;
    const size_t SZ_ATT = (size_t)BB * SEQ * SEQ;
    const size_t SZ_MV  = (size_t)BB * SEQ * MVD;

    size_t off = 0;
    float* embA = ws + off; off += SZ_EMB;
    float* embB = ws + off; off += SZ_EMB;
    float* XAF  = ws + off; off += SZ_X;
    float* XAB  = ws + off; off += SZ_X;
    float* XBF  = ws + off; off += SZ_X;
    float* XBB  = ws + off; off += SZ_X;
    float* cAf  = ws + off; off += SZ_C;
    float* cAb  = ws + off; off += SZ_C;
    float* cBf  = ws + off; off += SZ_C;
    float* cBb  = ws + off; off += SZ_C;
    float* nAf  = ws + off; off += SZ_N;
    float* nAb  = ws + off; off += SZ_N;
    float* nBf  = ws + off; off += SZ_N;
    float* nBb  = ws + off; off += SZ_N;
    float* attF = ws + off; off += SZ_ATT;
    float* attBk= ws + off; off += SZ_ATT;
    float* meanBf = ws + off; off += SZ_C;
    float* maxBf  = ws + off; off += SZ_C;
    float* meanBb = ws + off; off += SZ_C;
    float* maxBb  = ws + off; off += SZ_C;
    float* meanAf = ws + off; off += SZ_C;
    float* maxAf  = ws + off; off += SZ_C;
    float* meanAb = ws + off; off += SZ_C;
    float* maxAb  = ws + off; off += SZ_C;
    float* mvA  = ws + off; off += SZ_MV;
    float* mvB  = ws + off; off += SZ_MV;
    float* aoAf = ws + off; off += SZ_C;
    float* aoAb = ws + off; off += SZ_C;
    float* aoBf = ws + off; off += SZ_C;
    float* aoBb = ws + off; off += SZ_C;
    float* hcat = ws + off; off += (size_t)BB * HCAT;
    float* out1 = ws + off; off += (size_t)BB * FC1D;
    // reuse dead context-projection buffers for aggregation projections
    float* XGAF = XAF;
    float* XGAB = XAB;
    float* XGBF = XBF;
    float* XGBB = XBB;

    const int TB = 256;
    auto gemm = [&](const float* X, const float* W, const float* b1, const float* b2,
                    float* C, int M, int N, int K, int act) {
        int waves  = (M / 16) * (N / 16);
        int blocks = (waves + 7) / 8;   // 8 waves per 256-thread block
        k_gemm_bt_wmma<<<blocks, TB, 0, stream>>>(X, W, b1, b2, C, M, N, K, act);
    };

    // 1) embedding gather
    k_gather<<<(BB * SEQ * WDIM + TB - 1) / TB, TB, 0, stream>>>(Atok, emb, embA, BB * SEQ);
    k_gather<<<(BB * SEQ * WDIM + TB - 1) / TB, TB, 0, stream>>>(Btok, emb, embB, BB * SEQ);

    // 2) context input projections (all timesteps batched): (2048x300)@(300->512)
    gemm(embA, ctx_Wih_f, ctx_bih_f, ctx_bhh_f, XAF, BB * SEQ, GD, WDIM, 0);
    gemm(embA, ctx_Wih_b, ctx_bih_b, ctx_bhh_b, XAB, BB * SEQ, GD, WDIM, 0);
    gemm(embB, ctx_Wih_f, ctx_bih_f, ctx_bhh_f, XBF, BB * SEQ, GD, WDIM, 0);
    gemm(embB, ctx_Wih_b, ctx_bih_b, ctx_bhh_b, XBB, BB * SEQ, GD, WDIM, 0);

    // 3) context LSTM recurrences (4 jobs, 1 block each, TDM-prefetched)
    {
        LstmJobs cj;
        cj.j[0] = LstmJob{XAF, ctx_Whh_f, cAf, 0};
        cj.j[1] = LstmJob{XAB, ctx_Whh_b, cAb, 1};
        cj.j[2] = LstmJob{XBF, ctx_Whh_f, cBf, 0};
        cj.j[3] = LstmJob{XBB, ctx_Whh_b, cBb, 1};
        k_lstm4<<<4, TB, 0, stream>>>(cj);
    }

    // 4) norms + attention
    k_norm<<<(BB * SEQ + TB - 1) / TB, TB, 0, stream>>>(cAf, nAf);
    k_norm<<<(BB * SEQ + TB - 1) / TB, TB, 0, stream>>>(cAb, nAb);
    k_norm<<<(BB * SEQ + TB - 1) / TB, TB, 0, stream>>>(cBf, nBf);
    k_norm<<<(BB * SEQ + TB - 1) / TB, TB, 0, stream>>>(cBb, nBb);
    k_attention<<<(BB * SEQ * SEQ + TB - 1) / TB, TB, 0, stream>>>(cAf, cBf, nAf, nBf, attF);
    k_attention<<<(BB * SEQ * SEQ + TB - 1) / TB, TB, 0, stream>>>(cAb, cBb, nAb, nBb, attBk);

    // 5) attentive mean/max
    int cgrid = (BB * SEQ * HD + TB - 1) / TB;
    k_attn_combine<<<cgrid, TB, 0, stream>>>(attF,  cBf, meanBf, maxBf, 0);
    k_attn_combine<<<cgrid, TB, 0, stream>>>(attBk, cBb, meanBb, maxBb, 0);
    k_attn_combine<<<cgrid, TB, 0, stream>>>(attF,  cAf, meanAf, maxAf, 1);
    k_attn_combine<<<cgrid, TB, 0, stream>>>(attBk, cAb, meanAb, maxAb, 1);

    // 6) matching -> mvA / mvB (offsets: full,max,am,ax for f then b)
    int mgrid = (BB * SEQ * LP + TB - 1) / TB;
    const int BS = SEQ * HD;
    // mvA
    k_full_match<<<mgrid, TB, 0, stream>>>(cAf, cBf + (size_t)(SEQ - 1) * HD, BS, 0,  mp_w1, mvA, 0 * LP);
    k_maxpool_match<<<mgrid, TB, 0, stream>>>(cAf, cBf, mp_w3, mvA, 1 * LP);
    k_full_match<<<mgrid, TB, 0, stream>>>(cAf, meanBf, BS, HD, mp_w5, mvA, 2 * LP);
    k_full_match<<<mgrid, TB, 0, stream>>>(cAf, maxBf,  BS, HD, mp_w7, mvA, 3 * LP);
    k_full_match<<<mgrid, TB, 0, stream>>>(cAb, cBb, BS, 0, mp_w2, mvA, 4 * LP);
    k_maxpool_match<<<mgrid, TB, 0, stream>>>(cAb, cBb, mp_w4, mvA, 5 * LP);
    k_full_match<<<mgrid, TB, 0, stream>>>(cAb, meanBb, BS, HD, mp_w6, mvA, 6 * LP);
    k_full_match<<<mgrid, TB, 0, stream>>>(cAb, maxBb,  BS, HD, mp_w8, mvA, 7 * LP);
    // mvB (roles swapped; cosine maxpool is symmetric)
    k_full_match<<<mgrid, TB, 0, stream>>>(cBf, cAf + (size_t)(SEQ - 1) * HD, BS, 0,  mp_w1, mvB, 0 * LP);
    k_maxpool_match<<<mgrid, TB, 0, stream>>>(cBf, cAf, mp_w3, mvB, 1 * LP);
    k_full_match<<<mgrid, TB, 0, stream>>>(cBf, meanAf, BS, HD, mp_w5, mvB, 2 * LP);
    k_full_match<<<mgrid, TB, 0, stream>>>(cBf, maxAf,  BS, HD, mp_w7, mvB, 3 * LP);
    k_full_match<<<mgrid, TB, 0, stream>>>(cBb, cAb, BS, 0, mp_w2, mvB, 4 * LP);
    k_maxpool_match<<<mgrid, TB, 0, stream>>>(cBb, cAb, mp_w4, mvB, 5 * LP);
    k_full_match<<<mgrid, TB, 0, stream>>>(cBb, meanAb, BS, HD, mp_w6, mvB, 6 * LP);
    k_full_match<<<mgrid, TB, 0, stream>>>(cBb, maxAb,  BS, HD, mp_w8, mvB, 7 * LP);

    // 7) aggregation input projections: (2048x160)@(160->512)
    gemm(mvA, agg_Wih_f, agg_bih_f, agg_bhh_f, XGAF, BB * SEQ, GD, MVD, 0);
    gemm(mvA, agg_Wih_b, agg_bih_b, agg_bhh_b, XGAB, BB * SEQ, GD, MVD, 0);
    gemm(mvB, agg_Wih_f, agg_bih_f, agg_bhh_f, XGBF, BB * SEQ, GD, MVD, 0);
    gemm(mvB, agg_Wih_b, agg_bih_b, agg_bhh_b, XGBB, BB * SEQ, GD, MVD, 0);

    // 8) aggregation LSTM recurrences
    {
        LstmJobs aj;
        aj.j[0] = LstmJob{XGAF, agg_Whh_f, aoAf, 0};
        aj.j[1] = LstmJob{XGAB, agg_Whh_b, aoAb, 1};
        aj.j[2] = LstmJob{XGBF, agg_Whh_f, aoBf, 0};
        aj.j[3] = LstmJob{XGBB, agg_Whh_b, aoBb, 1};
        k_lstm4<<<4, TB, 0, stream>>>(aj);
    }

    // 9) concat final hiddens, fc1 (tanh), fc2 + softmax
    k_hcat<<<(BB * HCAT + TB - 1) / TB, TB, 0, stream>>>(aoAf, aoAb, aoBf, aoBb, hcat);
    gemm(hcat, fc1_W, fc1_b, nullptr, out1, BB, FC1D, HCAT, 1);
    k_fc2_softmax<<<1, 32, 0, stream>>>(out1, fc2_W, fc2_b, (float*)d_out);
}